// TemporalPredictorLite_16097537425493
// MI455X (gfx1250) — compile-verified
//
#include <hip/hip_runtime.h>
#include <hip/hip_bf16.h>

typedef float v2f __attribute__((ext_vector_type(2)));
typedef float v8f __attribute__((ext_vector_type(8)));

#define HSTR 257   // LDS row stride (floats) to avoid bank conflicts
#define PLANE 16384  // 128*128

// ---------------------------------------------------------------------------
// Kernel 1: mean over each (b,s,c) 128x128 plane. 402 MB streamed -> 24 KB.
// One block per plane, 256 threads, float4 loads (4 KB/iter coalesced).
// ---------------------------------------------------------------------------
__global__ __launch_bounds__(256) void tp_pool_kernel(
    const float* __restrict__ bev, float* __restrict__ pooled) {
  const long plane = blockIdx.x;                 // 0 .. 6143
  const float4* src = (const float4*)(bev + plane * (long)PLANE);
  const int tid = threadIdx.x;
  float sum = 0.f;
#pragma unroll
  for (int j = 0; j < 16; ++j) {
    float4 v = src[tid + j * 256];
    sum += (v.x + v.y) + (v.z + v.w);
  }
  // wave32 reduction
#pragma unroll
  for (int off = 16; off > 0; off >>= 1)
    sum += __shfl_down(sum, off, 32);
  __shared__ float wsum[8];
  const int lane = tid & 31, wv = tid >> 5;
  if (lane == 0) wsum[wv] = sum;
  __syncthreads();
  if (tid == 0) {
    float t = 0.f;
#pragma unroll
    for (int i = 0; i < 8; ++i) t += wsum[i];
    pooled[plane] = t * (1.0f / (float)PLANE);
  }
}

// ---------------------------------------------------------------------------
// Kernel 2: frame-replacement select + h0 + 6-step GRU, all via f32 WMMA.
// 1 block, 512 threads = 16 waves. Wave w owns columns [16w, 16w+16).
// ---------------------------------------------------------------------------
__device__ __forceinline__ float sigmf(float x) { return 1.f / (1.f + expf(-x)); }

__global__ __launch_bounds__(512) void tp_head_kernel(
    const float* __restrict__ pooled,            // [8][3][256] means
    const unsigned char* __restrict__ img_is_valid, // [8][3] bool
    const float* __restrict__ init_w, const float* __restrict__ init_b,
    const float* __restrict__ hist_w, const float* __restrict__ hist_b,
    const float* __restrict__ time_emb,          // [6][256]
    const float* __restrict__ w_ih,              // [768][256]
    const float* __restrict__ w_hh,              // [768][256]
    const float* __restrict__ b_ih, const float* __restrict__ b_hh,
    float* __restrict__ out)                     // [8][6][256]
{
  __shared__ float sh_h[16 * HSTR];              // hidden state, rows padded to 16
  __shared__ float sh_ab[2 * 16 * HSTR];         // phase1: pres|hist ; phase2: gi[6][768]
  float* sh_pres = sh_ab;
  float* sh_hist = sh_ab + 16 * HSTR;
  float* sh_gi   = sh_ab;                        // reused after h0

  const int tid  = threadIdx.x;
  const int wv   = tid >> 5;                     // 0..15
  const int lane = tid & 31;
  const int half = lane >> 4;                    // 0 | 1  (K sub-split of WMMA frag)
  const int lr   = lane & 15;                    // row / col within 16

  // zero padded A operands (rows 8..15 must be 0)
  for (int i = tid; i < 2 * 16 * HSTR; i += 512) sh_ab[i] = 0.f;
  __syncthreads();

  // ---- Phase A: invalid-frame replacement + valid-weighted history mean ----
  for (int i = tid; i < 8 * 256; i += 512) {
    const int b = i >> 8, c = i & 255;
    const bool v0 = img_is_valid[b * 3 + 0] != 0;
    const bool v1 = img_is_valid[b * 3 + 1] != 0;
    const bool v2 = img_is_valid[b * 3 + 2] != 0;
    const bool any_inv = (!v0) || (!v1) || (!v2);
    const int first = (!v0) ? 0 : ((!v1) ? 1 : ((!v2) ? 2 : 0));
    const int src = min(first + 1, 2);
    const float p0 = pooled[(b * 3 + 0) * 256 + c];
    const float p1 = pooled[(b * 3 + 1) * 256 + c];
    const float p2 = pooled[(b * 3 + 2) * 256 + c];
    const float ps = pooled[(b * 3 + src) * 256 + c];
    const float e0 = (any_inv && 0 <= first) ? ps : p0;
    const float e1 = (any_inv && 1 <= first) ? ps : p1;
    const float e2 = (any_inv && 2 <= first) ? ps : p2;
    const float h0 = v0 ? 1.f : 0.f, h1 = v1 ? 1.f : 0.f, h2 = v2 ? 1.f : 0.f;
    const float denom = fmaxf(h0 + h1 + h2, 1.0f);
    sh_hist[b * HSTR + c] = (h0 * e0 + h1 * e1 + h2 * e2) / denom;
    sh_pres[b * HSTR + c] = e2;   // present frame (post-replacement)
  }
  __syncthreads();

  // ---- Phase B: h0 = tanh(pres@init_w^T + hist@hist_w^T + init_b + hist_b) ----
  {
    v8f c = {};
    const int n0 = wv * 16;
    for (int k0 = 0; k0 < 256; k0 += 4) {
      const int ka = k0 + half * 2;
      v2f a, bf;
      a.x  = sh_pres[lr * HSTR + ka];
      a.y  = sh_pres[lr * HSTR + ka + 1];
      bf.x = init_w[(n0 + lr) * 256 + ka];       // B[k][n] = W[n][k]
      bf.y = init_w[(n0 + lr) * 256 + ka + 1];
      c = __builtin_amdgcn_wmma_f32_16x16x4_f32(false, a, false, bf, (short)0, c, false, false);
      a.x  = sh_hist[lr * HSTR + ka];
      a.y  = sh_hist[lr * HSTR + ka + 1];
      bf.x = hist_w[(n0 + lr) * 256 + ka];
      bf.y = hist_w[(n0 + lr) * 256 + ka + 1];
      c = __builtin_amdgcn_wmma_f32_16x16x4_f32(false, a, false, bf, (short)0, c, false, false);
    }
#pragma unroll
    for (int v = 0; v < 8; ++v) {
      const int m = v + half * 8;
      const int n = n0 + lr;
      sh_h[m * HSTR + n] = (m < 8) ? tanhf(c[v] + init_b[n] + hist_b[n]) : 0.f;
    }
  }
  __syncthreads();   // h0 done; pres/hist reads complete -> sh_ab reusable

  // ---- Phase C0: gi[t] = time_emb @ w_ih^T + b_ih (batch-independent) ----
  {
    const int nb = wv * 16;
    for (int g = 0; g < 3; ++g) {                // r / z / n gate blocks
      v8f c = {};
      const int n0 = g * 256 + nb;
      for (int k0 = 0; k0 < 256; k0 += 4) {
        const int ka = k0 + half * 2;
        v2f a, bf;
        a.x  = (lr < 6) ? time_emb[lr * 256 + ka] : 0.f;
        a.y  = (lr < 6) ? time_emb[lr * 256 + ka + 1] : 0.f;
        bf.x = w_ih[(n0 + lr) * 256 + ka];
        bf.y = w_ih[(n0 + lr) * 256 + ka + 1];
        c = __builtin_amdgcn_wmma_f32_16x16x4_f32(false, a, false, bf, (short)0, c, false, false);
      }
      if (half == 0) {
#pragma unroll
        for (int v = 0; v < 6; ++v) {            // only t = 0..5 rows are real
          const int n = n0 + lr;
          sh_gi[v * 768 + n] = c[v] + b_ih[n];
        }
      }
    }
  }
  __syncthreads();

  // ---- Phase C: GRU, 6 steps. Wave w fuses r/z/n for its 16 columns. ----
  const int nb = wv * 16;
  for (int t = 0; t < 6; ++t) {
    v8f cr = {}, cz = {}, cn = {};
    for (int k0 = 0; k0 < 256; k0 += 4) {
      const int ka = k0 + half * 2;
      v2f a, br, bz, bn;
      a.x  = sh_h[lr * HSTR + ka];
      a.y  = sh_h[lr * HSTR + ka + 1];
      br.x = w_hh[(nb + lr) * 256 + ka];         br.y = w_hh[(nb + lr) * 256 + ka + 1];
      bz.x = w_hh[(256 + nb + lr) * 256 + ka];   bz.y = w_hh[(256 + nb + lr) * 256 + ka + 1];
      bn.x = w_hh[(512 + nb + lr) * 256 + ka];   bn.y = w_hh[(512 + nb + lr) * 256 + ka + 1];
      cr = __builtin_amdgcn_wmma_f32_16x16x4_f32(false, a, false, br, (short)0, cr, false, false);
      cz = __builtin_amdgcn_wmma_f32_16x16x4_f32(false, a, false, bz, (short)0, cz, false, false);
      cn = __builtin_amdgcn_wmma_f32_16x16x4_f32(false, a, false, bn, (short)0, cn, false, false);
    }
    __syncthreads();   // all waves done reading sh_h for this step

    float hnew[8];
#pragma unroll
    for (int v = 0; v < 8; ++v) {
      const int m  = v + half * 8;
      const int nc = nb + lr;
      const float r = sigmf(sh_gi[t * 768 + nc]       + cr[v] + b_hh[nc]);
      const float z = sigmf(sh_gi[t * 768 + 256 + nc] + cz[v] + b_hh[256 + nc]);
      const float n = tanhf(sh_gi[t * 768 + 512 + nc] + r * (cn[v] + b_hh[512 + nc]));
      const float hprev = sh_h[m * HSTR + nc];   // own columns only
      hnew[v] = (1.f - z) * n + z * hprev;
    }
#pragma unroll
    for (int v = 0; v < 8; ++v) {
      const int m  = v + half * 8;
      const int nc = nb + lr;
      sh_h[m * HSTR + nc] = hnew[v];
      if (m < 8) out[(m * 6 + t) * 256 + nc] = hnew[v];
    }
    __syncthreads();
  }
}

// ---------------------------------------------------------------------------
extern "C" void kernel_launch(void* const* d_in, const int* in_sizes, int n_in,
                              void* d_out, int out_size, void* d_ws, size_t ws_size,
                              hipStream_t stream) {
  const float* bev          = (const float*)d_in[0];
  /* d_in[1] future_egomotion: dead — spatial mean is roll-invariant */
  const unsigned char* valid = (const unsigned char*)d_in[2];
  const float* init_w = (const float*)d_in[3];
  const float* init_b = (const float*)d_in[4];
  const float* hist_w = (const float*)d_in[5];
  const float* hist_b = (const float*)d_in[6];
  const float* time_emb = (const float*)d_in[7];
  const float* w_ih = (const float*)d_in[8];
  const float* w_hh = (const float*)d_in[9];
  const float* b_ih = (const float*)d_in[10];
  const float* b_hh = (const float*)d_in[11];
  float* out = (float*)d_out;
  float* pooled = (float*)d_ws;                  // [8][3][256] = 24 KB

  tp_pool_kernel<<<dim3(8 * 3 * 256), dim3(256), 0, stream>>>(bev, pooled);
  tp_head_kernel<<<dim3(1), dim3(512), 0, stream>>>(
      pooled, valid, init_w, init_b, hist_w, hist_b, time_emb,
      w_ih, w_hh, b_ih, b_hh, out);
}